// GraphAttnBlock_60550448939725
// MI455X (gfx1250) — compile-verified
//
#include <hip/hip_runtime.h>
#include <hip/hip_bf16.h>

#define HEADS 4
#define DCH   64
#define HD    (HEADS * DCH)   // 256

typedef __attribute__((ext_vector_type(2))) float v2f;
typedef __attribute__((ext_vector_type(8))) float v8f;

// Monotone uint encoding of float for atomicMax-based segment max.
__device__ __forceinline__ unsigned enc_f32(float f) {
  unsigned u = __float_as_uint(f);
  return (u & 0x80000000u) ? ~u : (u | 0x80000000u);
}
__device__ __forceinline__ float dec_f32(unsigned u) {
  unsigned b = (u & 0x80000000u) ? (u & 0x7FFFFFFFu) : ~u;
  return __uint_as_float(b);
}

// C[M,Ncols] = A[M,K] @ B[K,Ncols], fp32 WMMA. M%16==0, K%4==0, Ncols%64==0 or 16-tiled by grid.y*4 waves.
// One wave -> one 16x16 tile. blockDim=128 (4 waves): wave w covers col tile (blockIdx.y*4+w).
__global__ void gemm_wmma_f32(const float* __restrict__ A,
                              const float* __restrict__ B,
                              float* __restrict__ C,
                              int K, int Ncols) {
  const int lane = threadIdx.x & 31;
  const int wave = threadIdx.x >> 5;
  const int half = lane >> 4;      // lanes 16..31 -> 1
  const int lm   = lane & 15;
  const int row0 = blockIdx.x << 4;
  const int col0 = ((blockIdx.y << 2) + wave) << 4;
  const float* arow = A + (size_t)(row0 + lm) * K;
  v8f acc = {};
  for (int k0 = 0; k0 < K; k0 += 4) {
    const int ka = k0 + 2 * half;   // A lane layout: VGPR v holds K = k0 + 2*half + v
    v2f a, b;
    a.x = arow[ka];
    a.y = arow[ka + 1];
    b.x = B[(size_t)ka * Ncols + col0 + lm];         // B[k][n], n = lane%16
    b.y = B[(size_t)(ka + 1) * Ncols + col0 + lm];
    acc = __builtin_amdgcn_wmma_f32_16x16x4_f32(false, a, false, b,
                                                (short)0, acc, false, false);
  }
#pragma unroll
  for (int v = 0; v < 8; ++v)      // D: M = v + 8*half, N = lane%16
    C[(size_t)(row0 + v + 8 * half) * Ncols + col0 + lm] = acc[v];
}

__global__ void init_buffers(unsigned* __restrict__ menc, float* __restrict__ ssum,
                             float* __restrict__ accum, int n) {
  int idx = blockIdx.x * blockDim.x + threadIdx.x;
  if (idx < n * HEADS) { menc[idx] = 0u; ssum[idx] = 0.f; }
  if (idx < n * HD)    accum[idx] = 0.f;
}

// a_src[n,h] = dot(xw[n,h,:], att_src[h,:]);  a_dst likewise. One thread per (n,h).
__global__ void att_coef_kernel(const float* __restrict__ xw,
                                const float* __restrict__ att_src,
                                const float* __restrict__ att_dst,
                                float* __restrict__ a_src,
                                float* __restrict__ a_dst, int n) {
  int idx = blockIdx.x * blockDim.x + threadIdx.x;
  if (idx >= n * HEADS) return;
  int node = idx >> 2, h = idx & 3;
  const float4* xp = (const float4*)(xw + (size_t)node * HD + h * DCH);
  const float4* as = (const float4*)(att_src + h * DCH);
  const float4* ad = (const float4*)(att_dst + h * DCH);
  float ss = 0.f, sd = 0.f;
#pragma unroll
  for (int c = 0; c < DCH / 4; ++c) {
    float4 v = xp[c], s4 = as[c], d4 = ad[c];
    ss += v.x * s4.x + v.y * s4.y + v.z * s4.z + v.w * s4.w;
    sd += v.x * d4.x + v.y * d4.y + v.z * d4.z + v.w * d4.w;
  }
  a_src[idx] = ss;
  a_dst[idx] = sd;
}

__device__ __forceinline__ float edge_logit(const float* a_src, const float* a_dst,
                                            int s, int d, int h) {
  float l = a_src[s * HEADS + h] + a_dst[d * HEADS + h];
  return (l > 0.f) ? l : 0.2f * l;   // leaky_relu, slope 0.2
}

__global__ void edge_max_kernel(const int* __restrict__ src, const int* __restrict__ dst,
                                const float* __restrict__ a_src, const float* __restrict__ a_dst,
                                unsigned* __restrict__ menc, int E) {
  int idx = blockIdx.x * blockDim.x + threadIdx.x;
  if (idx >= E * HEADS) return;
  int e = idx >> 2, h = idx & 3;
  int s = src[e], d = dst[e];
  float l = edge_logit(a_src, a_dst, s, d, h);
  atomicMax(&menc[d * HEADS + h], enc_f32(l));   // global_atomic_max_u32, L2-resident
}

__global__ void edge_expsum_kernel(const int* __restrict__ src, const int* __restrict__ dst,
                                   const float* __restrict__ a_src, const float* __restrict__ a_dst,
                                   const unsigned* __restrict__ menc, float* __restrict__ ssum,
                                   float* __restrict__ exbuf, int E) {
  int idx = blockIdx.x * blockDim.x + threadIdx.x;
  if (idx >= E * HEADS) return;
  int e = idx >> 2, h = idx & 3;
  int s = src[e], d = dst[e];
  float l  = edge_logit(a_src, a_dst, s, d, h);
  float mm = dec_f32(menc[d * HEADS + h]);
  float ex = __expf(l - mm);
  exbuf[idx] = ex;
  atomicAdd(&ssum[d * HEADS + h], ex);           // global_atomic_add_f32
}

// One 256-thread block per edge: thread t handles channel t of the H*D=256 message.
__global__ void edge_aggregate_kernel(const int* __restrict__ src, const int* __restrict__ dst,
                                      const float* __restrict__ xw,
                                      const float* __restrict__ exbuf,
                                      const float* __restrict__ ssum,
                                      float* __restrict__ accum, int E) {
  int e = blockIdx.x;
  int t = threadIdx.x;        // 0..255; head = t>>6
  int h = t >> 6;
  int s = src[e], d = dst[e];
  float alpha = exbuf[(size_t)e * HEADS + h] / (ssum[d * HEADS + h] + 1e-16f);
  float v = xw[(size_t)s * HD + t] * alpha;      // coalesced 256-wide gather (L2-resident)
  atomicAdd(&accum[(size_t)d * HD + t], v);      // coalesced scatter-add at L2 atomic units
}

// out[n,c] = maybe_relu( mean_h accum[n,h,c] + bias[c] + lin[n,c] + lbias[c] )
__global__ void combine_kernel(const float* __restrict__ accum, const float* __restrict__ lin,
                               const float* __restrict__ bias, const float* __restrict__ lbias,
                               float* __restrict__ out, int n, int do_relu) {
  int idx = blockIdx.x * blockDim.x + threadIdx.x;
  if (idx >= n * DCH) return;
  int node = idx >> 6, c = idx & 63;
  const float* ap = accum + (size_t)node * HD;
  float mean = 0.25f * (ap[c] + ap[DCH + c] + ap[2 * DCH + c] + ap[3 * DCH + c]);
  float val = mean + bias[c] + lin[idx] + lbias[c];
  out[idx] = do_relu ? fmaxf(val, 0.f) : val;
}

extern "C" void kernel_launch(void* const* d_in, const int* in_sizes, int n_in,
                              void* d_out, int out_size, void* d_ws, size_t ws_size,
                              hipStream_t stream) {
  const float* x   = (const float*)d_in[0];
  const int*   ei  = (const int*)d_in[1];
  const float* W1  = (const float*)d_in[2];
  const float* as1 = (const float*)d_in[3];
  const float* ad1 = (const float*)d_in[4];
  const float* b1  = (const float*)d_in[5];
  const float* lW1 = (const float*)d_in[6];
  const float* lb1 = (const float*)d_in[7];
  const float* W2  = (const float*)d_in[8];
  const float* as2 = (const float*)d_in[9];
  const float* ad2 = (const float*)d_in[10];
  const float* b2  = (const float*)d_in[11];
  const float* lW2 = (const float*)d_in[12];
  const float* lb2 = (const float*)d_in[13];

  const int n = in_sizes[0] / DCH;     // 50000 (multiple of 16)
  const int E = in_sizes[1] / 2;       // 800000
  const int* srcI = ei;
  const int* dstI = ei + E;

  // Workspace carve-out (floats). Everything ~140 MB -> fits the 192 MB L2.
  float* ws = (float*)d_ws;
  size_t o = 0;
  float*    xw    = ws + o;            o += (size_t)n * HD;     // [N,256]
  float*    lin   = ws + o;            o += (size_t)n * DCH;    // [N,64]
  float*    asrc  = ws + o;            o += (size_t)n * HEADS;
  float*    adst  = ws + o;            o += (size_t)n * HEADS;
  unsigned* menc  = (unsigned*)(ws + o); o += (size_t)n * HEADS;
  float*    ssum  = ws + o;            o += (size_t)n * HEADS;
  float*    exbuf = ws + o;            o += (size_t)E * HEADS;  // [E,4]
  float*    accum = ws + o;            o += (size_t)n * HD;     // [N,256]
  float*    hbuf  = ws + o;            o += (size_t)n * DCH;    // [N,64]

  const dim3 gemmBlock(128);                       // 4 waves
  const dim3 gridXW(n / 16, HD / 64);              // (3125, 4): 256 output cols
  const dim3 gridLin(n / 16, DCH / 64);            // (3125, 1): 64 output cols
  const int nh = n * HEADS, eh = E * HEADS, nd = n * DCH;

  auto run_layer = [&](const float* X, const float* W, const float* as, const float* ad,
                       const float* b, const float* lW, const float* lb,
                       float* Y, int relu) {
    init_buffers<<<(n * HD + 255) / 256, 256, 0, stream>>>(menc, ssum, accum, n);
    gemm_wmma_f32<<<gridXW, gemmBlock, 0, stream>>>(X, W, xw, DCH, HD);
    gemm_wmma_f32<<<gridLin, gemmBlock, 0, stream>>>(X, lW, lin, DCH, DCH);
    att_coef_kernel<<<(nh + 255) / 256, 256, 0, stream>>>(xw, as, ad, asrc, adst, n);
    edge_max_kernel<<<(eh + 255) / 256, 256, 0, stream>>>(srcI, dstI, asrc, adst, menc, E);
    edge_expsum_kernel<<<(eh + 255) / 256, 256, 0, stream>>>(srcI, dstI, asrc, adst,
                                                             menc, ssum, exbuf, E);
    edge_aggregate_kernel<<<E, 256, 0, stream>>>(srcI, dstI, xw, exbuf, ssum, accum, E);
    combine_kernel<<<(nd + 255) / 256, 256, 0, stream>>>(accum, lin, b, lb, Y, n, relu);
  };

  run_layer(x,    W1, as1, ad1, b1, lW1, lb1, hbuf,           /*relu=*/1);
  run_layer(hbuf, W2, as2, ad2, b2, lW2, lb2, (float*)d_out,  /*relu=*/0);
}